// RNNLangModel_53979148976441
// MI455X (gfx1250) — compile-verified
//
#include <hip/hip_runtime.h>

// ---------------------------------------------------------------------------
// RNN language model, CDNA5 (gfx1250), wave32, V_WMMA_F32_16X16X4_F32 path.
//   Phase 1: fused embedding gather + input projection -> xprojT tiles (ws),
//            stored pre-permuted into phase-2 per-lane fragment order, with
//            BOTH biases (b_ih + b_hh) pre-folded in.
//   Phase 2: sequential scan on transposed state h^T (B operand); one wave
//            per SIMD, D->B re-layout = 6 shfl_xor + 4 selects, v_tanh_f32.
//   Phase 3: transposed head GEMM out^T = U_W @ h^T -> 2 x b128 stores/lane.
// ---------------------------------------------------------------------------

typedef __attribute__((ext_vector_type(2))) float v2f;
typedef __attribute__((ext_vector_type(8))) float v8f;

static constexpr int VOCAB = 9892;
static constexpr int EMB   = 100;
static constexpr int HID   = 10;
static constexpr int B     = 256;
static constexpr int T     = 2048;
static constexpr int BT16  = B / 16;                 // 16 batch tiles
static constexpr int NT    = (VOCAB + 15) / 16;      // 619 vocab tiles

__device__ __forceinline__ float fast_tanh(float x) {
#if defined(__gfx1250__) && __has_builtin(__builtin_amdgcn_tanhf)
    return __builtin_amdgcn_tanhf(x);
#elif defined(__gfx1250__) && __has_builtin(__builtin_amdgcn_tanh_f32)
    return __builtin_amdgcn_tanh_f32(x);
#else
    float r;
    // V_TANH_F32 is a gfx1250 TRANS op; trailing v_nop covers the TRANS
    // result-use hazard (1 independent op required).
    asm("v_tanh_f32 %0, %1\n\tv_nop" : "=v"(r) : "v"(x));
    return r;
#endif
}

// ---------------------------------------------------------------------------
// Phase 1: one wave = one 16x16 tile (16 batch rows at fixed t) x W_ih^T.
// EMB=100 -> 25 WMMA K=4 steps. D element (mb = v+8*half, nh = lane%16) is
// stored into fragment order for phase 2:
//   addr = tile*256 + (nh/8)*128 + half*64 + v*8 + (nh%8)
// so a phase-2 lane reads its whole v8f fragment as 8 contiguous floats.
// Both b_ih and b_hh are folded into the stored value.
// ---------------------------------------------------------------------------
__global__ __launch_bounds__(256) void rnn_xproj_kernel(
    const int* __restrict__ x, const float* __restrict__ emb,
    const float* __restrict__ W_ih, const float* __restrict__ b_ih,
    const float* __restrict__ b_hh, float* __restrict__ xprojT)
{
    const int lane = threadIdx.x & 31;
    const int tile = blockIdx.x * 8 + (threadIdx.x >> 5);   // 0 .. BT16*T-1
    const int btile = tile / T;
    const int t     = tile % T;
    const int row  = lane & 15;
    const int half = lane >> 4;

    // A rows: embedding rows of the 16 tokens of this tile
    const int b   = btile * 16 + row;
    const int tok = x[b * T + t];
    const float* erow = emb + (size_t)tok * EMB;

    // B columns: W_ih rows (N = hidden unit), branch-free masking past HID
    const int   n     = row;
    const int   nsafe = (n < HID) ? n : 0;
    const float nm    = (n < HID) ? 1.0f : 0.0f;
    const float* wrow = W_ih + nsafe * EMB;

    v8f c = {};
#pragma unroll
    for (int j = 0; j < 25; ++j) {                  // K = 100 = 25 * 4
        const int k = j * 4 + half * 2;
        v2f a, bb;
        a.x  = erow[k];
        a.y  = erow[k + 1];
        bb.x = wrow[k] * nm;
        bb.y = wrow[k + 1] * nm;
        c = __builtin_amdgcn_wmma_f32_16x16x4_f32(
                false, a, false, bb, (short)0, c, false, false);
    }

    // fold BOTH biases here so the scan's serial path has no bias add
    const float bias = (n < HID) ? (b_ih[n] + b_hh[n]) : 0.0f;
    float* base1 = xprojT + (size_t)tile * 256
                 + (n >> 3) * 128 + half * 64 + (n & 7);
#pragma unroll
    for (int v = 0; v < 8; ++v) base1[v * 8] = c[v] + bias;
}

// ---------------------------------------------------------------------------
// Phase 2: sequential scan, one wave per workgroup (one per SIMD), 16 total.
// State h^T (hid x batch) kept in D layout; per step:
//   c = xp^T tile (biases pre-folded), += W_hh x h^T via 3 independent
//   WMMAs, tanh. D -> B re-layout: 6 shfl_xor(16) + 4 selects (b2's high
//   half multiplies the masked-to-zero k=10,11 columns of aw[2]).
// ---------------------------------------------------------------------------
__global__ __launch_bounds__(32) void rnn_scan_kernel(
    const float* __restrict__ xprojT, const float* __restrict__ W_hh,
    float* __restrict__ hfinalT)
{
    const int lane  = threadIdx.x & 31;
    const int btile = blockIdx.x;                   // 0..15
    const int row   = lane & 15;
    const int half  = lane >> 4;

    // A fragments: W_hh (M = out hid = row, K = in hid), masked past HID
    v2f aw[3];
#pragma unroll
    for (int j = 0; j < 3; ++j) {
        const int k = j * 4 + half * 2;
        const bool m0 = (row < HID) && (k < HID);
        const bool m1 = (row < HID) && (k + 1 < HID);
        aw[j].x = W_hh[m0 ? (row * HID + k) : 0]     * (m0 ? 1.0f : 0.0f);
        aw[j].y = W_hh[m1 ? (row * HID + k + 1) : 0] * (m1 ? 1.0f : 0.0f);
    }

    const float* base = xprojT + (size_t)btile * T * 256;

    // Software-pipelined fragment loads: 2 x b128 per lane per step
    float4 x0 = *(const float4*)(base + lane * 8);
    float4 x1 = *(const float4*)(base + lane * 8 + 4);

    v8f h = {};                                     // h^T in D layout
    for (int t = 0; t < T; ++t) {
        const int tp = (t + 16 < T) ? (t + 16) : (T - 1);
        __builtin_prefetch(base + (size_t)tp * 256 + lane * 8, 0, 0);

        float4 n0 = x0, n1 = x1;
        if (t + 1 < T) {
            const float* nb = base + (size_t)(t + 1) * 256 + lane * 8;
            n0 = *(const float4*)nb;
            n1 = *(const float4*)(nb + 4);
        }

        // D -> B re-layout of h^T: B reg pair j holds K rows
        // {4j, 4j+1} (lanes 0-15) and {4j+2, 4j+3} (lanes 16-31).
        const float s0 = __shfl_xor(h[0], 16);
        const float s1 = __shfl_xor(h[1], 16);
        const float s2 = __shfl_xor(h[2], 16);
        const float s3 = __shfl_xor(h[3], 16);
        const float s6 = __shfl_xor(h[6], 16);
        const float s7 = __shfl_xor(h[7], 16);
        v2f b0, b1, b2;
        b0.x = half ? s2 : h[0];  b0.y = half ? s3 : h[1];
        b1.x = half ? s6 : h[4];  b1.y = half ? s7 : h[5];
        b2.x = s0;                b2.y = s1;   // K=10,11 hit zeroed A cols

        v8f c0 = { x0.x, x0.y, x0.z, x0.w, x1.x, x1.y, x1.z, x1.w };
        v8f z  = {};
        // three independent WMMAs: ~1 pipeline latency instead of 3
        c0     = __builtin_amdgcn_wmma_f32_16x16x4_f32(
                     false, aw[0], false, b0, (short)0, c0, false, false);
        v8f c1 = __builtin_amdgcn_wmma_f32_16x16x4_f32(
                     false, aw[1], false, b1, (short)0, z, false, false);
        v8f c2 = __builtin_amdgcn_wmma_f32_16x16x4_f32(
                     false, aw[2], false, b2, (short)0, z, false, false);

#pragma unroll
        for (int v = 0; v < 8; ++v)
            h[v] = fast_tanh((c0[v] + c1[v]) + c2[v]);

        x0 = n0; x1 = n1;
    }

    // Store h^T tile plainly: hfinalT[btile][m_hid][n_batch]
#pragma unroll
    for (int v = 0; v < 8; ++v)
        hfinalT[btile * 256 + (v + 8 * half) * 16 + row] = h[v];
}

// ---------------------------------------------------------------------------
// Phase 3: out^T tile = U_W(16xK) x h^T(Kx16). Each lane's 8 results are
// 8 consecutive vocab columns of one batch row -> 2 x b128 stores.
// ---------------------------------------------------------------------------
__global__ __launch_bounds__(256) void rnn_head_kernel(
    const float* __restrict__ hfinalT, const float* __restrict__ U_W,
    const float* __restrict__ U_b, float* __restrict__ out)
{
    const int lane = threadIdx.x & 31;
    const int wave = (blockIdx.x * 256 + threadIdx.x) >> 5;
    if (wave >= BT16 * NT) return;                  // wave-uniform exit
    const int btile = wave / NT;
    const int ntile = wave % NT;
    const int row  = lane & 15;
    const int half = lane >> 4;

    const int  vrow   = ntile * 16 + row;           // A row = vocab word
    const bool vvalid = (vrow < VOCAB);
    const float* hb = hfinalT + btile * 256;

    v8f c = {};
#pragma unroll
    for (int j = 0; j < 3; ++j) {
        const int k = j * 4 + half * 2;
        const bool m0 = vvalid && (k < HID);
        const bool m1 = vvalid && (k + 1 < HID);
        v2f a, bb;
        a.x  = U_W[m0 ? ((size_t)vrow * HID + k) : 0]     * (m0 ? 1.0f : 0.0f);
        a.y  = U_W[m1 ? ((size_t)vrow * HID + k + 1) : 0] * (m1 ? 1.0f : 0.0f);
        bb.x = hb[k * 16 + row];                    // h^T rows >= HID are 0
        bb.y = hb[(k + 1) * 16 + row];
        c = __builtin_amdgcn_wmma_f32_16x16x4_f32(
                false, a, false, bb, (short)0, c, false, false);
    }

    // D element (m_vocab = v+8*half, n_batch = lane%16)
    const int brow = btile * 16 + row;              // batch row
    const int col0 = ntile * 16 + 8 * half;         // first of 8 columns
    float* dst = out + (size_t)brow * VOCAB + col0;

    if ((ntile + 1) * 16 <= VOCAB) {                // uniform fast path
        float4 lo, hi;
        lo.x = c[0] + U_b[col0 + 0];  lo.y = c[1] + U_b[col0 + 1];
        lo.z = c[2] + U_b[col0 + 2];  lo.w = c[3] + U_b[col0 + 3];
        hi.x = c[4] + U_b[col0 + 4];  hi.y = c[5] + U_b[col0 + 5];
        hi.z = c[6] + U_b[col0 + 6];  hi.w = c[7] + U_b[col0 + 7];
        *(float4*)dst       = lo;
        *(float4*)(dst + 4) = hi;
    } else {                                        // last vocab tile
#pragma unroll
        for (int v = 0; v < 8; ++v) {
            const int col = col0 + v;
            if (col < VOCAB) dst[v] = c[v] + U_b[col];
        }
    }
}

// ---------------------------------------------------------------------------
extern "C" void kernel_launch(void* const* d_in, const int* in_sizes, int n_in,
                              void* d_out, int out_size, void* d_ws, size_t ws_size,
                              hipStream_t stream) {
    const int*   x    = (const int*)d_in[0];
    const float* emb  = (const float*)d_in[1];
    const float* W_ih = (const float*)d_in[2];
    const float* W_hh = (const float*)d_in[3];
    const float* b_ih = (const float*)d_in[4];
    const float* b_hh = (const float*)d_in[5];
    const float* U_W  = (const float*)d_in[6];
    const float* U_b  = (const float*)d_in[7];
    float* out = (float*)d_out;

    // ws layout: xprojT tiles (16*2048 tiles * 1KB = 32 MB) + hfinalT (16 KB)
    float* xprojT  = (float*)d_ws;
    float* hfinalT = xprojT + (size_t)BT16 * T * 256;

    rnn_xproj_kernel<<<(BT16 * T) / 8, 256, 0, stream>>>(x, emb, W_ih, b_ih, b_hh, xprojT);
    rnn_scan_kernel<<<BT16, 32, 0, stream>>>(xprojT, W_hh, hfinalT);
    rnn_head_kernel<<<(BT16 * NT + 7) / 8, 256, 0, stream>>>(hfinalT, U_W, U_b, out);
}